// ModelCore_7310034338468
// MI455X (gfx1250) — compile-verified
//
#include <hip/hip_runtime.h>
#include <math.h>

// Sizes from the reference
#define B_    512
#define SIG_  512
#define HID_  1024
#define EDGE_ 512
#define SLOTS_ 256
#define MEM_  512
#define IOC_  256
#define GRP_  32

typedef __attribute__((ext_vector_type(2))) float v2f;
typedef __attribute__((ext_vector_type(8))) float v8f;

// ---------------------------------------------------------------------------
// WMMA f32 GEMM:  C[M,N] = act(A[M,K] @ W[N,K]^T + bias[N] (+ addend[M,N]))
// act: 0=none, 1=silu, 2=sigmoid
// Requires M%64==0, N%128==0, K%16==0 (true for all call sites here).
// Block: 256 threads = 8 waves; block tile 64(M) x 128(N).
// Each wave owns a 32x32 subtile = 4 independent 16x16 WMMA accumulator
// chains (interleaved to hide WMMA->WMMA RAW latency).
// Double-buffered LDS + register prefetch: one barrier per K-chunk, global
// loads for chunk k+1 in flight while chunk k's 16 WMMAs execute.
// ---------------------------------------------------------------------------
#define BM 64
#define BN 128
#define KT 16

__global__ __launch_bounds__(256) void gemm_act_kernel(
    const float* __restrict__ A, const float* __restrict__ W,
    const float* __restrict__ bias, const float* __restrict__ addend,
    float* __restrict__ C, int M, int N, int K, int act)
{
    __shared__ float As[2][BM][KT + 1];   // 8704 B  (pad: stride 17, conflict-free)
    __shared__ float Ws[2][BN][KT + 1];   // 17408 B

    const int tid  = threadIdx.x;
    const int wave = tid >> 5;
    const int lane = tid & 31;
    const int m0 = blockIdx.y * BM;
    const int n0 = blockIdx.x * BN;
    const int mW = (wave >> 2) << 5;     // 0 or 32
    const int nW = (wave & 3) << 5;      // 0,32,64,96

    const int mrow  = lane & 15;         // A-row / B-col within a 16x16 tile
    const int khalf = (lane >> 4) << 1;  // K sub-offset: 0 or 2

    // global staging addresses (advance by KT per chunk)
    const int ar = tid >> 2;             // 0..63   A row, float4
    const int ac = (tid & 3) << 2;       // 0,4,8,12
    const int wr = tid >> 1;             // 0..127  W row, 2x float4
    const int wc = (tid & 1) << 3;       // 0 or 8
    const float* Aptr = A + (size_t)(m0 + ar) * K + ac;
    const float* Wptr = W + (size_t)(n0 + wr) * K + wc;

    v8f acc00 = {}, acc10 = {}, acc01 = {}, acc11 = {};

    const int nchunks = K / KT;

    // stage chunk 0
    float4 a_pf  = *(const float4*)(Aptr);
    float4 w_pf0 = *(const float4*)(Wptr);
    float4 w_pf1 = *(const float4*)(Wptr + 4);
    As[0][ar][ac]     = a_pf.x;  As[0][ar][ac + 1] = a_pf.y;
    As[0][ar][ac + 2] = a_pf.z;  As[0][ar][ac + 3] = a_pf.w;
    Ws[0][wr][wc]     = w_pf0.x; Ws[0][wr][wc + 1] = w_pf0.y;
    Ws[0][wr][wc + 2] = w_pf0.z; Ws[0][wr][wc + 3] = w_pf0.w;
    Ws[0][wr][wc + 4] = w_pf1.x; Ws[0][wr][wc + 5] = w_pf1.y;
    Ws[0][wr][wc + 6] = w_pf1.z; Ws[0][wr][wc + 7] = w_pf1.w;
    __syncthreads();

    for (int k = 0; k < nchunks; ++k) {
        const int cur  = k & 1;
        const int nxt  = cur ^ 1;
        const bool more = (k + 1) < nchunks;

        if (more) {   // issue next chunk's global loads before the math
            const float* Ap = Aptr + (size_t)(k + 1) * KT;
            const float* Wp = Wptr + (size_t)(k + 1) * KT;
            a_pf  = *(const float4*)(Ap);
            w_pf0 = *(const float4*)(Wp);
            w_pf1 = *(const float4*)(Wp + 4);
        }

        #pragma unroll
        for (int kk = 0; kk < KT; kk += 4) {
            v2f a0, a1, b0, b1;
            a0[0] = As[cur][mW + mrow][kk + khalf];
            a0[1] = As[cur][mW + mrow][kk + khalf + 1];
            a1[0] = As[cur][mW + 16 + mrow][kk + khalf];
            a1[1] = As[cur][mW + 16 + mrow][kk + khalf + 1];
            b0[0] = Ws[cur][nW + mrow][kk + khalf];
            b0[1] = Ws[cur][nW + mrow][kk + khalf + 1];
            b1[0] = Ws[cur][nW + 16 + mrow][kk + khalf];
            b1[1] = Ws[cur][nW + 16 + mrow][kk + khalf + 1];
            acc00 = __builtin_amdgcn_wmma_f32_16x16x4_f32(false, a0, false, b0, (short)0, acc00, false, false);
            acc10 = __builtin_amdgcn_wmma_f32_16x16x4_f32(false, a1, false, b0, (short)0, acc10, false, false);
            acc01 = __builtin_amdgcn_wmma_f32_16x16x4_f32(false, a0, false, b1, (short)0, acc01, false, false);
            acc11 = __builtin_amdgcn_wmma_f32_16x16x4_f32(false, a1, false, b1, (short)0, acc11, false, false);
        }

        if (more) {   // park the prefetched chunk in the other buffer
            As[nxt][ar][ac]     = a_pf.x;  As[nxt][ar][ac + 1] = a_pf.y;
            As[nxt][ar][ac + 2] = a_pf.z;  As[nxt][ar][ac + 3] = a_pf.w;
            Ws[nxt][wr][wc]     = w_pf0.x; Ws[nxt][wr][wc + 1] = w_pf0.y;
            Ws[nxt][wr][wc + 2] = w_pf0.z; Ws[nxt][wr][wc + 3] = w_pf0.w;
            Ws[nxt][wr][wc + 4] = w_pf1.x; Ws[nxt][wr][wc + 5] = w_pf1.y;
            Ws[nxt][wr][wc + 6] = w_pf1.z; Ws[nxt][wr][wc + 7] = w_pf1.w;
        }
        __syncthreads();
    }

    // Epilogue. C/D layout: lane holds N = lane&15, VGPR v holds M = v + 8*(lane>>4)
    const int nbase = n0 + nW + mrow;
    const int mbase = m0 + mW + ((lane >> 4) << 3);
    auto epi = [&](const v8f& a, int mt, int nt) {
        const int n = nbase + nt;
        const float bv = bias ? bias[n] : 0.0f;
        #pragma unroll
        for (int v = 0; v < 8; ++v) {
            const int m = mbase + mt + v;
            float x = a[v] + bv;
            if (addend) x += addend[(size_t)m * N + n];
            if (act == 1)      x = x / (1.0f + __expf(-x));      // silu
            else if (act == 2) x = 1.0f / (1.0f + __expf(-x));   // sigmoid
            C[(size_t)m * N + n] = x;
        }
    };
    epi(acc00, 0, 0); epi(acc10, 16, 0); epi(acc01, 0, 16); epi(acc11, 16, 16);
}

// ---------------------------------------------------------------------------
// energy[b] = relu(dot(latent[b,:HID], ew) + eb) ; one wave per row
// ---------------------------------------------------------------------------
__global__ __launch_bounds__(256) void energy_kernel(
    const float* __restrict__ latent, const float* __restrict__ ew,
    const float* __restrict__ eb, float* __restrict__ out)
{
    int wave = threadIdx.x >> 5, lane = threadIdx.x & 31;
    int row = blockIdx.x * 8 + wave;
    const float* lr = latent + (size_t)row * HID_;
    float acc = 0.0f;
    for (int j = lane; j < HID_; j += 32) acc += lr[j] * ew[j];
    for (int off = 16; off > 0; off >>= 1) acc += __shfl_xor(acc, off, 32);
    if (lane == 0) out[row] = fmaxf(acc + eb[0], 0.0f);
}

// ---------------------------------------------------------------------------
// Row softmax over SLOTS=256; one block (256 threads) per row.
// ---------------------------------------------------------------------------
__global__ __launch_bounds__(256) void softmax256_kernel(
    const float* __restrict__ logits, float* __restrict__ out)
{
    __shared__ float red[256];
    int b = blockIdx.x, s = threadIdx.x;
    float x = logits[(size_t)b * SLOTS_ + s];
    red[s] = x; __syncthreads();
    for (int off = 128; off > 0; off >>= 1) {
        if (s < off) red[s] = fmaxf(red[s], red[s + off]);
        __syncthreads();
    }
    float mx = red[0]; __syncthreads();
    float e = __expf(x - mx);
    red[s] = e; __syncthreads();
    for (int off = 128; off > 0; off >>= 1) {
        if (s < off) red[s] += red[s + off];
        __syncthreads();
    }
    out[(size_t)b * SLOTS_ + s] = e / red[0];
}

// ---------------------------------------------------------------------------
// Fused per-batch memory pipeline. One block per b, 256 threads.
//   read[b,d]    = sum_s w[s]*mem[b,s,d]
//   gate[b]      = sigmoid(dot(latent[b],wg[:HID]) + dot(read[b],wg[HID:]) + wgb)
//   new_mem[bsd] = (1-g*w[s])*mem[b,s,d] + g*w[s]*value[b,d]
// Pass 1 streams memory[b] (512KB) with prefetch hints, keeping it L2-hot.
// Pass 2 re-reads it with NT (last-use) loads and writes new_memory with NT
// stores (write-once data must not evict the L2 working set).
// ---------------------------------------------------------------------------
__global__ __launch_bounds__(256) void memory_kernel(
    const float* __restrict__ weights, const float* __restrict__ memory,
    const float* __restrict__ latent, const float* __restrict__ wg_w,
    const float* __restrict__ wg_b, const float* __restrict__ value,
    float* __restrict__ readb, float* __restrict__ new_memory)
{
    __shared__ float wsm[SLOTS_];
    __shared__ float red[256];
    const int b = blockIdx.x, tid = threadIdx.x;
    wsm[tid] = weights[(size_t)b * SLOTS_ + tid];
    __syncthreads();

    const v2f* mb = (const v2f*)(memory + (size_t)b * SLOTS_ * MEM_);
    float r0 = 0.0f, r1 = 0.0f;                 // d = 2*tid, 2*tid+1
    for (int s = 0; s < SLOTS_; ++s) {
        if (((tid & 15) == 0) && (s + 12 < SLOTS_))
            __builtin_prefetch((const void*)(mb + (s + 12) * (MEM_ / 2) + tid), 0, 1);
        v2f mv = mb[s * (MEM_ / 2) + tid];
        float w = wsm[s];
        r0 += w * mv.x; r1 += w * mv.y;
    }
    readb[(size_t)b * MEM_ + 2 * tid]     = r0;
    readb[(size_t)b * MEM_ + 2 * tid + 1] = r1;

    // gate
    const float* lr = latent + (size_t)b * HID_;
    float acc = lr[tid]       * wg_w[tid]
              + lr[tid + 256] * wg_w[tid + 256]
              + lr[tid + 512] * wg_w[tid + 512]
              + lr[tid + 768] * wg_w[tid + 768]
              + r0 * wg_w[HID_ + 2 * tid]
              + r1 * wg_w[HID_ + 2 * tid + 1];
    red[tid] = acc; __syncthreads();
    for (int off = 128; off > 0; off >>= 1) {
        if (tid < off) red[tid] += red[tid + off];
        __syncthreads();
    }
    const float gate = 1.0f / (1.0f + __expf(-(red[0] + wg_b[0])));

    // memory update (second pass: NT/last-use in, NT out)
    const v2f val = ((const v2f*)(value + (size_t)b * MEM_))[tid];
    v2f* nm = (v2f*)(new_memory + (size_t)b * SLOTS_ * MEM_);
    for (int s = 0; s < SLOTS_; ++s) {
        float gw = gate * wsm[s];
        v2f mv = __builtin_nontemporal_load(&mb[s * (MEM_ / 2) + tid]);
        v2f o;
        o.x = (1.0f - gw) * mv.x + gw * val.x;
        o.y = (1.0f - gw) * mv.y + gw * val.y;
        __builtin_nontemporal_store(o, &nm[s * (MEM_ / 2) + tid]);
    }
}

// ---------------------------------------------------------------------------
// cat[b,:] = [refined[b,:EDGE], remap[b,:GRP]]   (K=544 input for a1 GEMM)
// ---------------------------------------------------------------------------
__global__ __launch_bounds__(256) void concat_kernel(
    const float* __restrict__ refined, const float* __restrict__ remap,
    float* __restrict__ cat)
{
    int b = blockIdx.x;
    for (int j = threadIdx.x; j < EDGE_ + GRP_; j += 256)
        cat[(size_t)b * (EDGE_ + GRP_) + j] =
            (j < EDGE_) ? refined[(size_t)b * EDGE_ + j]
                        : remap[(size_t)b * GRP_ + (j - EDGE_)];
}

// ---------------------------------------------------------------------------
// io[b,c] = sum_e softmax_e(base_map[c,e] + adapt[b,e]) * refined[b,e]
// One block per b; each wave owns output channels c (wave32 reductions).
// Never materializes the [B,IOC,EDGE] tensor (would be 256MB).
// ---------------------------------------------------------------------------
__global__ __launch_bounds__(256) void io_kernel(
    const float* __restrict__ base_map, const float* __restrict__ adapt,
    const float* __restrict__ refined, float* __restrict__ io)
{
    __shared__ float ad[EDGE_];
    __shared__ float rf[EDGE_];
    const int b = blockIdx.x, tid = threadIdx.x;
    ad[tid]       = adapt[(size_t)b * EDGE_ + tid];
    ad[tid + 256] = adapt[(size_t)b * EDGE_ + tid + 256];
    rf[tid]       = refined[(size_t)b * EDGE_ + tid];
    rf[tid + 256] = refined[(size_t)b * EDGE_ + tid + 256];
    __syncthreads();

    const int wave = tid >> 5, lane = tid & 31;
    for (int c = wave; c < IOC_; c += 8) {
        const float* bm = base_map + (size_t)c * EDGE_;
        float mv[16];
        float mx = -INFINITY;
        #pragma unroll
        for (int j = 0; j < 16; ++j) {
            int e = j * 32 + lane;
            mv[j] = bm[e] + ad[e];
            mx = fmaxf(mx, mv[j]);
        }
        for (int off = 16; off > 0; off >>= 1) mx = fmaxf(mx, __shfl_xor(mx, off, 32));
        float sum = 0.0f, dot = 0.0f;
        #pragma unroll
        for (int j = 0; j < 16; ++j) {
            float e = __expf(mv[j] - mx);
            sum += e;
            dot += e * rf[j * 32 + lane];
        }
        for (int off = 16; off > 0; off >>= 1) {
            sum += __shfl_xor(sum, off, 32);
            dot += __shfl_xor(dot, off, 32);
        }
        if (lane == 0) io[(size_t)b * IOC_ + c] = dot / sum;
    }
}

// ---------------------------------------------------------------------------
extern "C" void kernel_launch(void* const* d_in, const int* in_sizes, int n_in,
                              void* d_out, int out_size, void* d_ws, size_t ws_size,
                              hipStream_t stream)
{
    (void)in_sizes; (void)n_in; (void)out_size; (void)ws_size;
    const float* signal     = (const float*)d_in[0];
    const float* memory     = (const float*)d_in[1];
    const float* remap      = (const float*)d_in[2];
    const float* enc_w1     = (const float*)d_in[3];
    const float* enc_b1     = (const float*)d_in[4];
    const float* enc_w2     = (const float*)d_in[5];
    const float* enc_b2     = (const float*)d_in[6];
    const float* edge_w     = (const float*)d_in[7];
    const float* edge_b     = (const float*)d_in[8];
    const float* energy_w   = (const float*)d_in[9];
    const float* energy_b   = (const float*)d_in[10];
    const float* rq_w       = (const float*)d_in[11];
    const float* rq_b       = (const float*)d_in[12];
    const float* wg_w       = (const float*)d_in[13];
    const float* wg_b       = (const float*)d_in[14];
    const float* wv_w       = (const float*)d_in[15];
    const float* wv_b       = (const float*)d_in[16];
    const float* memory_key = (const float*)d_in[17];
    const float* m2e_w      = (const float*)d_in[18];
    const float* m2e_b      = (const float*)d_in[19];
    const float* base_map   = (const float*)d_in[20];
    const float* a1_w       = (const float*)d_in[21];
    const float* a1_b       = (const float*)d_in[22];
    const float* a2_w       = (const float*)d_in[23];
    const float* a2_b       = (const float*)d_in[24];

    // Output layout (flat, return order)
    float* latent_o  = (float*)d_out;                    // [B,HID]
    float* refined_o = latent_o + (size_t)B_ * HID_;     // [B,EDGE]
    float* energy_o  = refined_o + (size_t)B_ * EDGE_;   // [B,1]
    float* io_o      = energy_o + B_;                    // [B,IOC]
    float* newmem_o  = io_o + (size_t)B_ * IOC_;         // [B,SLOTS,MEM]
    float* weights_o = newmem_o + (size_t)B_ * SLOTS_ * MEM_; // [B,SLOTS]

    // Workspace layout (floats)
    float* ws        = (float*)d_ws;
    float* lat1      = ws;                                   // [B,HID]
    float* readiness = lat1 + (size_t)B_ * HID_;             // [B,EDGE]
    float* query     = readiness + (size_t)B_ * EDGE_;       // [B,MEM]
    float* value     = query + (size_t)B_ * MEM_;            // [B,MEM]
    float* logits    = value + (size_t)B_ * MEM_;            // [B,SLOTS]
    float* readb     = logits + (size_t)B_ * SLOTS_;         // [B,MEM]
    float* catb      = readb + (size_t)B_ * MEM_;            // [B,EDGE+GRP]
    float* t1        = catb + (size_t)B_ * (EDGE_ + GRP_);   // [B,EDGE]
    float* adapt     = t1 + (size_t)B_ * EDGE_;              // [B,EDGE]

    const dim3 blk(256);
    auto grid = [](int M, int N) { return dim3((unsigned)(N / BN), (unsigned)(M / BM)); };

    // Encoder
    gemm_act_kernel<<<grid(B_, HID_), blk, 0, stream>>>(
        signal, enc_w1, enc_b1, nullptr, lat1, B_, HID_, SIG_, 1);
    gemm_act_kernel<<<grid(B_, HID_), blk, 0, stream>>>(
        lat1, enc_w2, enc_b2, nullptr, latent_o, B_, HID_, HID_, 1);

    // Heads off latent
    gemm_act_kernel<<<grid(B_, EDGE_), blk, 0, stream>>>(
        latent_o, edge_w, edge_b, nullptr, readiness, B_, EDGE_, HID_, 2);
    gemm_act_kernel<<<grid(B_, MEM_), blk, 0, stream>>>(
        latent_o, rq_w, rq_b, nullptr, query, B_, MEM_, HID_, 0);
    gemm_act_kernel<<<grid(B_, MEM_), blk, 0, stream>>>(
        latent_o, wv_w, wv_b, nullptr, value, B_, MEM_, HID_, 0);
    energy_kernel<<<dim3(B_ / 8), blk, 0, stream>>>(latent_o, energy_w, energy_b, energy_o);

    // Attention over memory slots
    gemm_act_kernel<<<grid(B_, SLOTS_), blk, 0, stream>>>(
        query, memory_key, nullptr, nullptr, logits, B_, SLOTS_, MEM_, 0);
    softmax256_kernel<<<dim3(B_), blk, 0, stream>>>(logits, weights_o);

    // Fused read / gate / memory-update (HBM-bound core)
    memory_kernel<<<dim3(B_), blk, 0, stream>>>(
        weights_o, memory, latent_o, wg_w, wg_b, value, readb, newmem_o);

    // refined = sigmoid(readiness + read @ m2e^T + b)
    gemm_act_kernel<<<grid(B_, EDGE_), blk, 0, stream>>>(
        readb, m2e_w, m2e_b, readiness, refined_o, B_, EDGE_, MEM_, 2);

    // Router
    concat_kernel<<<dim3(B_), blk, 0, stream>>>(refined_o, remap, catb);
    gemm_act_kernel<<<grid(B_, EDGE_), blk, 0, stream>>>(
        catb, a1_w, a1_b, nullptr, t1, B_, EDGE_, EDGE_ + GRP_, 1);
    gemm_act_kernel<<<grid(B_, EDGE_), blk, 0, stream>>>(
        t1, a2_w, a2_b, nullptr, adapt, B_, EDGE_, EDGE_, 0);

    // io = softmax(base_map + adapt) . refined
    io_kernel<<<dim3(B_), blk, 0, stream>>>(base_map, adapt, refined_o, io_o);
}